// GCN_ten_89704686944358
// MI455X (gfx1250) — compile-verified
//
#include <hip/hip_runtime.h>
#include <math.h>

// ---------------------------------------------------------------------------
// GCN (PyG GCNConv x10) for MI455X / gfx1250.
//   deg/dis/norm precompute  -> fp32 WMMA GEMM per layer (16-wide hidden dim)
//   -> self-loop+bias init   -> edge scatter (float atomics, 16 lanes/edge)
//   -> final 16x40 GEMM + log_softmax fused per node.
// Aggregation is linear, so the last layer aggregates BEFORE the 16->40 GEMM,
// keeping every scatter exactly 16 floats wide (minimum HBM traffic).
// ---------------------------------------------------------------------------

typedef float v2f __attribute__((ext_vector_type(2)));
typedef float v8f __attribute__((ext_vector_type(8)));

#define TPB 256

// ---- degree / normalization precompute ------------------------------------

__global__ void k_fill1(float* __restrict__ p, int n) {
  int i = blockIdx.x * blockDim.x + threadIdx.x;
  if (i < n) p[i] = 1.0f;                       // self-loop contributes 1
}

__global__ void k_deg(const long long* __restrict__ dst, float* __restrict__ deg, int E) {
  int e = blockIdx.x * blockDim.x + threadIdx.x;
  if (e < E) atomicAdd(&deg[(int)dst[e]], 1.0f);
}

__global__ void k_dis(float* __restrict__ deg, int n) {   // in-place rsqrt
  int i = blockIdx.x * blockDim.x + threadIdx.x;
  if (i < n) deg[i] = rsqrtf(deg[i]);           // deg >= 1 always
}

__global__ void k_edge_prep(const long long* __restrict__ src64,
                            const long long* __restrict__ dst64,
                            const float* __restrict__ dis,
                            int* __restrict__ src32, int* __restrict__ dst32,
                            float* __restrict__ norm, int E) {
  int e = blockIdx.x * blockDim.x + threadIdx.x;
  if (e >= E) return;
  int s = (int)src64[e];
  int d = (int)dst64[e];
  src32[e] = s;
  dst32[e] = d;
  norm[e] = dis[s] * dis[d];
}

// ---- fp32 WMMA GEMM: out[N x 16] = act(A[N x K]) @ W[K x 16] --------------
// One wave = one 16-row tile. V_WMMA_F32_16X16X4_F32, K reduced 4 at a time.
// A layout (16x4/lane): row = lane&15, K-pair koff = 2*(lane>>4)  (ISA 7.12.2)
// B layout (4x16):      col = lane&15, rows koff, koff+1
// D layout:             VGPR r -> row r + 8*(lane>>4), col = lane&15

__global__ void k_gemm16(const float* __restrict__ A, const float* __restrict__ W,
                         float* __restrict__ out, int K, int relu_in, int ntiles) {
  int wave = (int)((blockIdx.x * blockDim.x + threadIdx.x) >> 5);
  int lane = threadIdx.x & 31;
  if (wave >= ntiles) return;                   // wave-uniform: EXEC all-1s below

  int row  = wave * 16 + (lane & 15);
  int koff = (lane >> 4) << 1;                  // 0 or 2
  int col  = lane & 15;

  const float* __restrict__ arow = A + (size_t)row * K;

  v8f c = {0.f, 0.f, 0.f, 0.f, 0.f, 0.f, 0.f, 0.f};
  for (int k0 = 0; k0 < K; k0 += 4) {
    float a0 = arow[k0 + koff];
    float a1 = arow[k0 + koff + 1];
    if (relu_in) { a0 = fmaxf(a0, 0.f); a1 = fmaxf(a1, 0.f); }
    v2f a; a.x = a0; a.y = a1;
    v2f b; b.x = W[(k0 + koff) * 16 + col];
           b.y = W[(k0 + koff + 1) * 16 + col];
    c = __builtin_amdgcn_wmma_f32_16x16x4_f32(
        /*neg_a=*/false, a, /*neg_b=*/false, b,
        /*c_mod=*/(short)0, c, /*reuse_a=*/false, /*reuse_b=*/false);
  }

#pragma unroll
  for (int r = 0; r < 8; r++) {
    int m = wave * 16 + r + ((lane >> 4) << 3);
    out[(size_t)m * 16 + col] = c[r];
  }
}

// ---- accumulator init: agg = bias + dis^2 * act(t)  (self-loop folded) ----

__global__ void k_agg_init(const float* __restrict__ t, const float* __restrict__ dis,
                           const float* __restrict__ bias, int use_bias, int relu_in,
                           float* __restrict__ agg, int total /* N*16 */) {
  int i = blockIdx.x * blockDim.x + threadIdx.x;
  if (i >= total) return;
  int n = i >> 4;
  int j = i & 15;
  float v = t[i];
  if (relu_in) v = fmaxf(v, 0.f);
  float d = dis[n];
  float r = d * d * v;
  if (use_bias) r += bias[j];
  agg[i] = r;
}

// ---- edge scatter: agg[dst] += norm * act(t[src]), 16 lanes per edge ------

__global__ void k_scatter(const float* __restrict__ t,
                          const int* __restrict__ src, const int* __restrict__ dst,
                          const float* __restrict__ norm, int relu_in,
                          float* __restrict__ agg, int total /* E*16 */) {
  int i = blockIdx.x * blockDim.x + threadIdx.x;
  if (i >= total) return;
  int e = i >> 4;
  int j = i & 15;
  int s = src[e];
  int d = dst[e];
  float v = t[(size_t)s * 16 + j];
  if (relu_in) v = fmaxf(v, 0.f);
  atomicAdd(&agg[(size_t)d * 16 + j], norm[e] * v);
}

// ---- final: logits = h[16] @ w_out[16x40] + b_out, then log_softmax -------

__global__ void k_final(const float* __restrict__ h, const float* __restrict__ w,
                        const float* __restrict__ b, float* __restrict__ out, int N) {
  __shared__ float sw[16 * 40];
  __shared__ float sb[40];
  __shared__ float sh[TPB * 16];

  int t = threadIdx.x;
  for (int i = t; i < 16 * 40; i += blockDim.x) sw[i] = w[i];
  for (int i = t; i < 40; i += blockDim.x)      sb[i] = b[i];

  int base = blockIdx.x * blockDim.x;           // first node of this block
  for (int i = t; i < TPB * 16; i += blockDim.x) {
    size_t idx = (size_t)base * 16 + i;         // coalesced stage of h rows
    if (idx < (size_t)N * 16) sh[i] = h[idx];
  }
  __syncthreads();

  int n = base + t;
  if (n >= N) return;

  float hn[16];
#pragma unroll
  for (int k = 0; k < 16; k++) hn[k] = sh[t * 16 + k];

  float logits[40];
  float m = -INFINITY;
#pragma unroll
  for (int j = 0; j < 40; j++) {
    float acc = sb[j];
#pragma unroll
    for (int k = 0; k < 16; k++) acc += hn[k] * sw[k * 40 + j];
    logits[j] = acc;
    m = fmaxf(m, acc);
  }
  float sum = 0.f;
#pragma unroll
  for (int j = 0; j < 40; j++) sum += expf(logits[j] - m);
  float lse = logf(sum);
  float* __restrict__ o = out + (size_t)n * 40;
#pragma unroll
  for (int j = 0; j < 40; j++) o[j] = logits[j] - m - lse;
}

// ---------------------------------------------------------------------------

extern "C" void kernel_launch(void* const* d_in, const int* in_sizes, int n_in,
                              void* d_out, int out_size, void* d_ws, size_t ws_size,
                              hipStream_t stream) {
  const int F = 128, H = 16, C = 40;
  const int N = in_sizes[0] / F;        // 200000
  const int E = in_sizes[7] / 2;        // 6400000

  const float*     x      = (const float*)d_in[0];
  const float*     w_in   = (const float*)d_in[1];
  const float*     b_in   = (const float*)d_in[2];
  const float*     w_hid  = (const float*)d_in[3];   // [8][16][16]
  const float*     b_hid  = (const float*)d_in[4];   // [8][16]
  const float*     w_out  = (const float*)d_in[5];   // [16][40]
  const float*     b_out  = (const float*)d_in[6];   // [40]
  const long long* edges  = (const long long*)d_in[7];
  const long long* src64  = edges;
  const long long* dst64  = edges + E;

  // workspace carve-out (256B aligned)
  char* ws = (char*)d_ws;
  size_t off = 0;
  auto carve = [&](size_t bytes) -> void* {
    void* p = ws + off;
    off = (off + bytes + 255) & ~(size_t)255;
    return p;
  };
  float* dis   = (float*)carve((size_t)N * 4);        // deg, then rsqrt in place
  float* normv = (float*)carve((size_t)E * 4);
  int*   src32 = (int*)  carve((size_t)E * 4);
  int*   dst32 = (int*)  carve((size_t)E * 4);
  float* tbuf  = (float*)carve((size_t)N * H * 4);    // GEMM output / final agg
  float* hbuf  = (float*)carve((size_t)N * H * 4);    // layer activations
  (void)ws_size; (void)n_in; (void)out_size;

  const int ntiles   = N / 16;                        // 12500 (exact)
  const int gemm_blk = (ntiles * 32 + TPB - 1) / TPB;
  const int nh_blk   = (N * H + TPB - 1) / TPB;
  const int eh_blk   = (E * H + TPB - 1) / TPB;
  const int n_blk    = (N + TPB - 1) / TPB;
  const int e_blk    = (E + TPB - 1) / TPB;

  // ---- normalization precompute (once) ----
  k_fill1    <<<n_blk, TPB, 0, stream>>>(dis, N);
  k_deg      <<<e_blk, TPB, 0, stream>>>(dst64, dis, E);
  k_dis      <<<n_blk, TPB, 0, stream>>>(dis, N);
  k_edge_prep<<<e_blk, TPB, 0, stream>>>(src64, dst64, dis, src32, dst32, normv, E);

  // ---- layers 1..9: z = agg(act(h) @ W) + b ;  h' = relu(z) (relu deferred)
  const float* cur = x;
  int K = F, relu_in = 0;
  for (int layer = 0; layer < 9; layer++) {
    const float* Wl = (layer == 0) ? w_in : (w_hid + (size_t)(layer - 1) * H * H);
    const float* bl = (layer == 0) ? b_in : (b_hid + (size_t)(layer - 1) * H);
    k_gemm16  <<<gemm_blk, TPB, 0, stream>>>(cur, Wl, tbuf, K, relu_in, ntiles);
    k_agg_init<<<nh_blk,   TPB, 0, stream>>>(tbuf, dis, bl, 1, 0, hbuf, N * H);
    k_scatter <<<eh_blk,   TPB, 0, stream>>>(tbuf, src32, dst32, normv, 0, hbuf, E * H);
    cur = hbuf; K = H; relu_in = 1;
  }

  // ---- layer 10: aggregate relu(z9) first (linearity), then 16->40 GEMM ----
  k_agg_init<<<nh_blk, TPB, 0, stream>>>(hbuf, dis, nullptr, 0, 1, tbuf, N * H);
  k_scatter <<<eh_blk, TPB, 0, stream>>>(hbuf, src32, dst32, normv, 1, tbuf, E * H);
  k_final   <<<n_blk,  TPB, 0, stream>>>(tbuf, w_out, b_out, (float*)d_out, N);
  (void)C;
}